// MHAttention_19310172963166
// MI455X (gfx1250) — compile-verified
//
#include <hip/hip_runtime.h>

// CDNA5 / gfx1250 implementation of causal prefix-mean:
//   out[b,t,:] = cumsum(x, axis=1)[b,t,:] / (t+1),  x: (8, 4096, 1024) f32
//
// Memory-bound (268 MB min traffic -> ~11.5 us @ 23.3 TB/s). Strategy:
//   pass 1: per-chunk partial sums (streams x from HBM, warms 192MB L2)
//   pass 2: tiny exclusive scan of 32 chunk-sums per column
//   pass 3: intra-chunk cumsum via tril(ones) matmul on V_WMMA_F32_16X16X4_F32
//           (keeps f32 precision; carry rides the WMMA C accumulator),
//           x re-read served by L2, non-temporal output stores.

typedef __attribute__((ext_vector_type(2))) float v2f;
typedef __attribute__((ext_vector_type(8))) float v8f;

constexpr int Bn    = 8;
constexpr int Tn    = 4096;
constexpr int Cn    = 1024;
constexpr int NCH   = 32;            // chunks along T
constexpr int CHUNK = Tn / NCH;      // 128
constexpr int TILES = CHUNK / 16;    // 8 WMMA tiles of 16 t-rows per chunk

// ---------------------------------------------------------------------------
// Kernel 1: partial sum of each (b, chunk) over its 128 t-rows, all 1024 ch.
// 256 threads/block, each owning a float4 (4 channels). Fully coalesced.
// ---------------------------------------------------------------------------
__global__ void chunk_sum_kernel(const float* __restrict__ x,
                                 float* __restrict__ part) {
  const int blk = blockIdx.x;            // b*NCH + ch
  const int b   = blk / NCH;
  const int ch  = blk % NCH;
  const int tid = threadIdx.x;           // 0..255 -> float4 slot across C
  const float4* x4 = (const float4*)x;
  size_t base = (size_t)(b * Tn + ch * CHUNK) * (Cn / 4) + tid;
  float4 acc = {0.f, 0.f, 0.f, 0.f};
#pragma unroll 8
  for (int t = 0; t < CHUNK; ++t) {
    float4 v = x4[base + (size_t)t * (Cn / 4)];
    acc.x += v.x; acc.y += v.y; acc.z += v.z; acc.w += v.w;
  }
  ((float4*)part)[(size_t)blk * (Cn / 4) + tid] = acc;
}

// ---------------------------------------------------------------------------
// Kernel 2: exclusive scan of the 32 chunk sums for each of the 8192 columns.
// ---------------------------------------------------------------------------
__global__ void chunk_scan_kernel(const float* __restrict__ part,
                                  float* __restrict__ excl) {
  const int c  = blockIdx.x * blockDim.x + threadIdx.x;  // 0..Bn*Cn-1
  const int b  = c / Cn;
  const int cc = c % Cn;
  float run = 0.f;
  for (int ch = 0; ch < NCH; ++ch) {
    size_t idx = (size_t)(b * NCH + ch) * Cn + cc;
    float v = part[idx];
    excl[idx] = run;
    run += v;
  }
}

// ---------------------------------------------------------------------------
// Kernel 3: one wave owns (b, chunk, 16-channel strip). For each 16x16 tile:
//   D = L @ X + Carry  with L = tril(ones(16,16)), done as 4 chained
//   v_wmma_f32_16x16x4_f32 (K=4 slices). D[m,n] = carry[n] + cumsum_t.
// Carry for next tile = D[15,:] (VGPR7, hi half) broadcast via shuffle.
// ---------------------------------------------------------------------------
__global__ void cummean_wmma_kernel(const float* __restrict__ x,
                                    const float* __restrict__ excl,
                                    float* __restrict__ out) {
  const int wid  = blockIdx.x * (blockDim.x >> 5) + (threadIdx.x >> 5);
  const int lane = threadIdx.x & 31;
  const int cs   = wid & 63;           // channel strip: 64 strips of 16
  const int ch   = (wid >> 6) & (NCH - 1);
  const int b    = wid >> 11;
  const int n    = lane & 15;          // N index (and M index for A operand)
  const int hi   = lane >> 4;          // half-wave selector
  const int c0   = cs * 16;

  // A operand = tril(ones) 16x16, split into four 16x4 K-slices.
  // A layout (f32 16x4): lanes 0-15 & 16-31 both carry M=lane&15;
  //   VGPR0 holds K = 4j + 2*hi, VGPR1 holds K = 4j + 2*hi + 1.
  v2f a[4];
#pragma unroll
  for (int j = 0; j < 4; ++j) {
    int k0 = 4 * j + 2 * hi;
    a[j].x = (k0     <= n) ? 1.f : 0.f;
    a[j].y = (k0 + 1 <= n) ? 1.f : 0.f;
  }

  // Initial carry: exclusive chunk-prefix for this column.
  float carry = excl[(size_t)(b * NCH + ch) * Cn + c0 + n];

  const int t0 = ch * CHUNK;
  for (int tile = 0; tile < TILES; ++tile) {
    const int tb = t0 + tile * 16;

    // B operand (f32 4x16 per slice): VGPR0 holds row K = 4j + 2*hi,
    // VGPR1 row K+1; lanes index N. Rows are t, cols are channels.
    v2f bb[4];
#pragma unroll
    for (int j = 0; j < 4; ++j) {
      int r = tb + 4 * j + 2 * hi;
      size_t idx = (size_t)(b * Tn + r) * Cn + c0 + n;
      bb[j].x = x[idx];
      bb[j].y = x[idx + Cn];
    }

    // C matrix: uniform carry[n] in every row -> same scalar in all 8 VGPRs.
    v8f acc;
#pragma unroll
    for (int i = 0; i < 8; ++i) acc[i] = carry;

    // D = sum_j A_j x B_j + C : four chained 16x16x4 f32 WMMAs.
#pragma unroll
    for (int j = 0; j < 4; ++j) {
      acc = __builtin_amdgcn_wmma_f32_16x16x4_f32(
          /*neg_a=*/false, a[j], /*neg_b=*/false, bb[j],
          /*c_mod=*/(short)0, acc, /*reuse_a=*/false, /*reuse_b=*/false);
    }

    // D layout: VGPR p -> row m = p + 8*hi. Divide by (t+1) and store NT
    // so output traffic doesn't evict x from L2.
#pragma unroll
    for (int p = 0; p < 8; ++p) {
      int t = tb + p + 8 * hi;
      size_t oidx = (size_t)(b * Tn + t) * Cn + c0 + n;
      float r = acc[p] * __builtin_amdgcn_rcpf((float)(t + 1));
      __builtin_nontemporal_store(r, &out[oidx]);
    }

    // New carry = D[15, n], held in VGPR7 of lane 16+n.
    carry = __shfl(acc[7], 16 + n, 32);
  }
}

extern "C" void kernel_launch(void* const* d_in, const int* in_sizes, int n_in,
                              void* d_out, int out_size, void* d_ws, size_t ws_size,
                              hipStream_t stream) {
  const float* x = (const float*)d_in[0];
  float* out  = (float*)d_out;
  float* part = (float*)d_ws;                          // Bn*NCH*Cn f32 = 1 MB
  float* excl = part + (size_t)Bn * NCH * Cn;          // another 1 MB

  // Pass 1: 256 blocks (b x chunk), 256 threads, float4 per thread.
  chunk_sum_kernel<<<Bn * NCH, 256, 0, stream>>>(x, part);
  // Pass 2: 8192 columns / 256 = 32 blocks.
  chunk_scan_kernel<<<(Bn * Cn) / 256, 256, 0, stream>>>(part, excl);
  // Pass 3: 16384 waves, 4 waves per 128-thread block.
  cummean_wmma_kernel<<<(Bn * NCH * (Cn / 16)) / 4, 128, 0, stream>>>(x, excl, out);
}